// LNPBlock_19172734009545
// MI455X (gfx1250) — compile-verified
//
#include <hip/hip_runtime.h>
#include <math.h>

typedef float v2f __attribute__((ext_vector_type(2)));
typedef float v8f __attribute__((ext_vector_type(8)));

#define B_   2
#define L_   385
#define N_   384     // tokens after cls
#define C_   128
#define C2_  256
#define EPSV 1e-5f
#define LDS_STRIDE 260   // 256 + 4 pad -> banks 4m / 4m+2, conflict-free b64 loads

// ---------------------------------------------------------------------------
// Kernel 1: per-(b,c) statistics.
//   s[b,c]      = sqrt(Var_j F[b,j,c] + eps)
//   scale0[b,c] = gamma[c] / s[b,c]
//   econst[b,c] = beta[c] + sum_j softmax_j(a)_j * a_j,  a_j = scale0*F[b,j,c]
// Also copies the cls-token row. grid = (B), block = (C) -> coalesced over c.
// ---------------------------------------------------------------------------
__global__ __launch_bounds__(C_) void lnp_stats_kernel(
    const float* __restrict__ x, const float* __restrict__ gamma,
    const float* __restrict__ beta, float* __restrict__ out,
    float* __restrict__ scale0, float* __restrict__ econst)
{
    const int b = blockIdx.x;
    const int c = threadIdx.x;
    const size_t base = (size_t)b * L_ * C_;
    const float* __restrict__ F = x + base + C_;   // skip cls row

    // cls token pass-through
    out[base + c] = x[base + c];

    float sum = 0.f;
    for (int j = 0; j < N_; ++j) sum += F[j * C_ + c];
    const float mean = sum * (1.0f / N_);

    float ss = 0.f;
    for (int j = 0; j < N_; ++j) { float d = F[j * C_ + c] - mean; ss += d * d; }
    const float var = ss * (1.0f / N_);             // jnp.var, ddof=0
    const float inv = 1.0f / sqrtf(var + EPSV);
    const float sc  = gamma[c] * inv;

    float mx = -3.0e38f;
    for (int j = 0; j < N_; ++j) mx = fmaxf(mx, sc * F[j * C_ + c]);

    float se = 0.f, sa = 0.f;
    for (int j = 0; j < N_; ++j) {
        float a = sc * F[j * C_ + c];
        float e = __expf(a - mx);
        se += e; sa += a * e;
    }
    scale0[b * C_ + c] = sc;
    econst[b * C_ + c] = sa / se + beta[c];
}

// ---------------------------------------------------------------------------
// Kernel 2: tokens = Fc_C @ W^T + bias via V_WMMA_F32_16X16X4_F32 (full fp32).
// One block per 16-row output tile (48 tiles over M = B*N = 768; 384%16==0 so
// tiles never cross the batch boundary). 256 threads = 8 wave32; wave w owns
// output columns [16w, 16w+16). K = 2C = 256 -> 64 WMMA steps of K=4.
// Fc_C tile (16 x 256) is reconstructed on the fly into LDS from F + stats.
// ---------------------------------------------------------------------------
__global__ __launch_bounds__(256) void lnp_gemm_kernel(
    const float* __restrict__ x, const float* __restrict__ gamma,
    const float* __restrict__ beta, const float* __restrict__ W,
    const float* __restrict__ bias, const float* __restrict__ scale0,
    const float* __restrict__ econst, float* __restrict__ out)
{
    __shared__ float ldsA[16 * LDS_STRIDE];

    const int row0 = blockIdx.x * 16;     // global row in [0, 768)
    const int b    = row0 / N_;
    const int i0   = row0 % N_;
    const int t    = threadIdx.x;         // doubles as column c' in staging

    // ---- Stage Fc_C[16 rows][256 cols] into LDS (thread t = column c') ----
    {
        const int  cp = t;
        const int  c  = cp & (C_ - 1);
        const bool lo = (cp < C_);
        const float sc = lo ? scale0[b * C_ + cp] : 0.f;
        const float ec = lo ? econst[b * C_ + cp] : 0.f;
        const float g1 = lo ? 0.f : gamma[cp];
        const float b1 = lo ? 0.f : beta[cp];
        const float* __restrict__ Fb = x + (size_t)b * L_ * C_ + C_;
        #pragma unroll
        for (int m = 0; m < 16; ++m) {
            float Fv = Fb[(size_t)(i0 + m) * C_ + c];
            ldsA[m * LDS_STRIDE + cp] = lo ? (ec - sc * Fv) : (g1 * Fv + b1);
        }
    }
    __syncthreads();

    const int wave  = t >> 5;
    const int lane  = t & 31;
    const int nbase = wave * 16;
    const int half  = (lane < 16) ? 0 : 2;      // K sub-pair this half-wave owns
    const int mrow  = lane & 15;                 // A: M index for this lane
    const int ncol  = nbase + (lane & 15);       // B/D: N (output channel)

    // Seed accumulator with bias (per-lane constant across the 8 D rows).
    v8f acc;
    const float bv = bias[ncol];
    #pragma unroll
    for (int q = 0; q < 8; ++q) acc[q] = bv;

    const float* __restrict__ wrow = W + (size_t)ncol * C2_;  // B[k,n] = W[n,c']

    #pragma unroll 4
    for (int kt = 0; kt < 64; ++kt) {
        const int c0 = kt * 4 + half;
        v2f a  = *(const v2f*)&ldsA[mrow * LDS_STRIDE + c0];  // 8B aligned, no bank conflict
        v2f bb = *(const v2f*)&wrow[c0];
        // D = A(16x4) * B(4x16) + C   (f32, round-to-nearest-even)
        acc = __builtin_amdgcn_wmma_f32_16x16x4_f32(
                  /*neg_a=*/false, a, /*neg_b=*/false, bb,
                  /*c_mod=*/(short)0, acc, /*reuse_a=*/false, /*reuse_b=*/false);
    }

    // D layout: VGPR v -> (M=v, lanes 0-15) and (M=v+8, lanes 16-31), N = ncol.
    const int mofs = (lane < 16) ? 0 : 8;
    float* __restrict__ outp = out + (size_t)b * L_ * C_ + (size_t)(1 + i0) * C_;
    #pragma unroll
    for (int v = 0; v < 8; ++v) {
        outp[(size_t)(v + mofs) * C_ + ncol] = acc[v];
    }
}

// ---------------------------------------------------------------------------
extern "C" void kernel_launch(void* const* d_in, const int* in_sizes, int n_in,
                              void* d_out, int out_size, void* d_ws, size_t ws_size,
                              hipStream_t stream) {
    const float* x     = (const float*)d_in[0];   // (2,385,128)
    const float* gamma = (const float*)d_in[1];   // (1,1,1,256)
    const float* beta  = (const float*)d_in[2];   // (1,1,1,256)
    const float* W     = (const float*)d_in[3];   // (128,256)
    const float* bias  = (const float*)d_in[4];   // (128,)
    float* out = (float*)d_out;                   // (2,385,128)

    float* scale0 = (float*)d_ws;                 // [B_*C_]
    float* econst = scale0 + B_ * C_;             // [B_*C_]  (2 KB total)

    lnp_stats_kernel<<<B_, C_, 0, stream>>>(x, gamma, beta, out, scale0, econst);
    lnp_gemm_kernel<<<(B_ * N_) / 16, 256, 0, stream>>>(
        x, gamma, beta, W, bias, scale0, econst, out);
}